// DeformableConvBlock_45835890983325
// MI455X (gfx1250) — compile-verified
//
#include <hip/hip_runtime.h>
#include <hip/hip_bf16.h>

typedef __attribute__((ext_vector_type(2))) float v2f;
typedef __attribute__((ext_vector_type(8))) float v8f;

// Problem constants (from setup_inputs): B=4, Cin=64, Cout=128, H=W=128
#define CIN   64
#define COUT  128
#define HH    128
#define WW    128
#define NB    4
#define KTAP  9
#define KDIM  (CIN * KTAP)   // 576
#define TN    64             // pixels per block (4 N-subtiles of 16)
#define VSTRIDE 72           // panel row stride: 2*72 mod 64 = 16 -> both K-half
                             // lane groups hit disjoint banks for all subtiles

#define WMMA_F32(A, B, C) \
  __builtin_amdgcn_wmma_f32_16x16x4_f32(false, (A), false, (B), (short)0, (C), false, false)

// ---------------------------------------------------------------------------
// Kernel 0: transpose main_w [o][c][k] -> wT[kk = k*64+c][m] so consumer
// A-fragment loads are lane-coalesced (consecutive m = consecutive addresses).
// ---------------------------------------------------------------------------
__global__ __launch_bounds__(256) void transpose_w_kernel(
    const float* __restrict__ mw, float* __restrict__ wT) {
  int i = blockIdx.x * 256 + threadIdx.x;  // 128*64*9 = 73728
  if (i >= COUT * CIN * KTAP) return;
  int k = i % 9;
  int c = (i / 9) % CIN;
  int m = i / (9 * CIN);
  wT[(size_t)(k * CIN + c) * COUT + m] = mw[i];
}

// ---------------------------------------------------------------------------
// Kernel 1: offset conv (3x3, Cin=64 -> 18), one thread per output pixel.
// Weights (18*576 floats = 40.5 KB) staged in LDS once per block.
// ---------------------------------------------------------------------------
__global__ __launch_bounds__(256) void offset_conv_kernel(
    const float* __restrict__ x, const float* __restrict__ ow,
    const float* __restrict__ ob, float* __restrict__ offs) {
  __shared__ float lds_w[KDIM * 18];  // layout: [(c*9+k)*18 + o]

  for (int f = threadIdx.x; f < KDIM * 18; f += 256) {
    int o  = f % 18;
    int ck = f / 18;
    int k  = ck % 9;
    int c  = ck / 9;
    lds_w[f] = ow[(o * CIN + c) * 9 + k];
  }
  __syncthreads();

  int idx = blockIdx.x * blockDim.x + threadIdx.x;  // 0 .. B*H*W-1
  int w = idx & (WW - 1);
  int h = (idx >> 7) & (HH - 1);
  int b = idx >> 14;

  float acc[18];
#pragma unroll
  for (int o = 0; o < 18; ++o) acc[o] = ob[o];

  for (int kh = 0; kh < 3; ++kh) {
    int y = h + kh - 1;
    if (y < 0 || y >= HH) continue;
    for (int kw = 0; kw < 3; ++kw) {
      int xx = w + kw - 1;
      if (xx < 0 || xx >= WW) continue;
      int k = kh * 3 + kw;
      const float* xp = x + ((size_t)(b * CIN) * HH + y) * WW + xx;
      for (int c = 0; c < CIN; ++c) {
        float xv = xp[(size_t)c * HH * WW];
        const float* wl = &lds_w[(c * 9 + k) * 18];
#pragma unroll
        for (int o = 0; o < 18; ++o) acc[o] += xv * wl[o];
      }
    }
  }

  float* op = offs + ((size_t)(b * 18) * HH + h) * WW + w;
#pragma unroll
  for (int o = 0; o < 18; ++o) op[(size_t)o * HH * WW] = acc[o];
}

// ---------------------------------------------------------------------------
// Producer staging: bilinear-sample one tap's 64ch x 64px panel into LDS.
// 128 producer threads: p = pid&63 (pixel), cg = pid>>6 (32 channels each).
// ---------------------------------------------------------------------------
__device__ __forceinline__ void stage_tap(
    float* __restrict__ panel, const float* __restrict__ xb,
    const float* __restrict__ offs, int b, int h, int w0, int pid, int tap) {
  int p    = pid & (TN - 1);
  int cg   = pid >> 6;          // 0..1
  int wpix = w0 + p;

  float dy = offs[(((size_t)b * 18 + 2 * tap) * HH + h) * WW + wpix];
  float dx = offs[(((size_t)b * 18 + 2 * tap + 1) * HH + h) * WW + wpix];
  int ky = tap / 3, kx = tap % 3;
  float py = (float)(h - 1 + ky) + dy;
  float px = (float)(wpix - 1 + kx) + dx;
  float fy = floorf(py), fx = floorf(px);
  int y0 = (int)fy, x0 = (int)fx;
  int y1 = y0 + 1, x1 = x0 + 1;
  float wy = py - fy, wx = px - fx;
  // validity folded into bilinear weights (zero-pad semantics)
  float vy0 = (y0 >= 0 && y0 < HH) ? 1.f : 0.f;
  float vy1 = (y1 >= 0 && y1 < HH) ? 1.f : 0.f;
  float vx0 = (x0 >= 0 && x0 < WW) ? 1.f : 0.f;
  float vx1 = (x1 >= 0 && x1 < WW) ? 1.f : 0.f;
  float m00 = (1.f - wy) * (1.f - wx) * vy0 * vx0;
  float m01 = (1.f - wy) * wx * vy0 * vx1;
  float m10 = wy * (1.f - wx) * vy1 * vx0;
  float m11 = wy * wx * vy1 * vx1;
  int yc0 = min(max(y0, 0), HH - 1), yc1 = min(max(y1, 0), HH - 1);
  int xc0 = min(max(x0, 0), WW - 1), xc1 = min(max(x1, 0), WW - 1);
  int i00 = yc0 * WW + xc0, i01 = yc0 * WW + xc1;
  int i10 = yc1 * WW + xc0, i11 = yc1 * WW + xc1;

#pragma unroll
  for (int j = 0; j < 32; ++j) {
    int c = cg * 32 + j;
    const float* xc_ = xb + (size_t)c * HH * WW;
    float v = m00 * xc_[i00] + m01 * xc_[i01] + m10 * xc_[i10] + m11 * xc_[i11];
    panel[c * VSTRIDE + p] = v;
  }
}

// ---------------------------------------------------------------------------
// Kernel 2: wave-specialized deformable GEMM.
// Waves 0-3 consume: each owns 2 Cout tiles x 4 pixel subtiles (8 acc v8f);
// per K-step: 2 coalesced A-frags (wT) + 4 shared B-frags -> 8 WMMAs.
// Waves 4-7 produce: double-buffered LDS panel (2 x 64 x 72 f32 = 36.9 KB),
// staging tap k+1 while tap k is consumed; one barrier per tap.
// ---------------------------------------------------------------------------
__global__ __launch_bounds__(256) void deform_gemm_kernel(
    const float* __restrict__ x, const float* __restrict__ offs,
    const float* __restrict__ wT, const float* __restrict__ mb,
    float* __restrict__ out) {
  __shared__ float Vs[2][CIN * VSTRIDE];  // 2*64*72*4 = 36864 B

  int t    = threadIdx.x;
  int tile = blockIdx.x;           // 1024 tiles
  int w0   = (tile & 1) * TN;      // 2 tiles per image row
  int h    = (tile >> 1) & (HH - 1);
  int b    = tile >> 8;

  int wave = t >> 5;               // 0..3 consume, 4..7 produce
  int lane = t & 31;
  bool isConsumer = (wave < 4);

  const float* xb = x + (size_t)b * CIN * HH * WW;

  // consumer fragment indexing (f32 16x16x4 layouts)
  int m0base = wave * 32;          // consumer wave owns M tiles [m0base, m0base+16]
  int lm = lane & 15;              // A row / B col / D col
  int lh = lane >> 4;              // K-half selector

  v8f c00 = {}, c01 = {}, c02 = {}, c03 = {};
  v8f c10 = {}, c11 = {}, c12 = {}, c13 = {};

  if (!isConsumer) stage_tap(Vs[0], xb, offs, b, h, w0, t - 128, 0);
  __syncthreads();

  for (int tap = 0; tap < KTAP; ++tap) {
    if (isConsumer) {
      const float* panel = Vs[tap & 1];
      const float* wtap  = wT + (size_t)tap * CIN * COUT;
#pragma unroll 2
      for (int ks = 0; ks < CIN / 4; ++ks) {
        v2f a0, a1, b0, b1, b2, b3;
#pragma unroll
        for (int j = 0; j < 2; ++j) {
          int kl = ks * 4 + lh * 2 + j;               // local K index in tap
          a0[j] = wtap[(size_t)kl * COUT + m0base + lm];        // coalesced
          a1[j] = wtap[(size_t)kl * COUT + m0base + 16 + lm];
          const float* vr = &panel[kl * VSTRIDE + lm];
          b0[j] = vr[0];
          b1[j] = vr[16];
          b2[j] = vr[32];
          b3[j] = vr[48];
        }
        c00 = WMMA_F32(a0, b0, c00);
        c01 = WMMA_F32(a0, b1, c01);
        c02 = WMMA_F32(a0, b2, c02);
        c03 = WMMA_F32(a0, b3, c03);
        c10 = WMMA_F32(a1, b0, c10);
        c11 = WMMA_F32(a1, b1, c11);
        c12 = WMMA_F32(a1, b2, c12);
        c13 = WMMA_F32(a1, b3, c13);
      }
    } else if (tap + 1 < KTAP) {
      stage_tap(Vs[(tap + 1) & 1], xb, offs, b, h, w0, t - 128, tap + 1);
    }
    __syncthreads();
  }

  // ---- epilogue: bias + store (C/D layout: vgpr r -> M = r + 8*lh) ----
  if (isConsumer) {
#pragma unroll
    for (int r = 0; r < 8; ++r) {
      int m0 = m0base + r + 8 * lh;
      int m1 = m0 + 16;
      float bias0 = mb[m0], bias1 = mb[m1];
      float* o0 = out + (((size_t)b * COUT + m0) * HH + h) * WW + w0 + lm;
      float* o1 = out + (((size_t)b * COUT + m1) * HH + h) * WW + w0 + lm;
      o0[0]  = c00[r] + bias0;
      o0[16] = c01[r] + bias0;
      o0[32] = c02[r] + bias0;
      o0[48] = c03[r] + bias0;
      o1[0]  = c10[r] + bias1;
      o1[16] = c11[r] + bias1;
      o1[32] = c12[r] + bias1;
      o1[48] = c13[r] + bias1;
    }
  }
}

extern "C" void kernel_launch(void* const* d_in, const int* in_sizes, int n_in,
                              void* d_out, int out_size, void* d_ws, size_t ws_size,
                              hipStream_t stream) {
  const float* x  = (const float*)d_in[0];
  const float* ow = (const float*)d_in[1];
  const float* ob = (const float*)d_in[2];
  const float* mw = (const float*)d_in[3];
  const float* mb = (const float*)d_in[4];
  float* out  = (float*)d_out;
  float* offs = (float*)d_ws;                          // B*18*H*W f32 = 4.7 MB
  float* wT   = (float*)d_ws + (size_t)NB * 18 * HH * WW;  // 576*128 f32 = 288 KB

  // Kernel 0: weight transpose (runs concurrently-independent of kernel 1 deps)
  transpose_w_kernel<<<(COUT * CIN * KTAP + 255) / 256, 256, 0, stream>>>(mw, wT);

  // Kernel 1: one thread per pixel, B*H*W = 65536 threads
  offset_conv_kernel<<<(NB * HH * WW) / 256, 256, 0, stream>>>(x, ow, ob, offs);

  // Kernel 2: one block per 64-pixel tile: 1024 blocks x 256 threads
  deform_gemm_kernel<<<(NB * HH * WW) / TN, 256, 0, stream>>>(x, offs, wT, mb, out);
}